// Entity_Classify_49821620633803
// MI455X (gfx1250) — compile-verified
//
#include <hip/hip_runtime.h>
#include <hip/hip_bf16.h>

// Problem constants (from the reference)
constexpr int N   = 50000;
constexpr int D   = 128;
constexpr int R   = 8;
constexpr int E   = 80000;
constexpr int B   = 4;
constexpr int NF  = 4;
constexpr int V   = 1000;
constexpr int OUT = 16;

typedef __attribute__((ext_vector_type(2))) float v2f;
typedef __attribute__((ext_vector_type(8))) float v8f;
typedef int v4i __attribute__((vector_size(16)));   // param type of async-LDS builtins

// gfx1250 async Global->LDS copy (ASYNCcnt-tracked), guarded for portability
#if __has_builtin(__builtin_amdgcn_global_load_async_to_lds_b128) && \
    __has_builtin(__builtin_amdgcn_s_wait_asynccnt)
#define USE_ASYNC_LDS 1
#else
#define USE_ASYNC_LDS 0
#endif

// ---------------------------------------------------------------------------
// Fold weights, stored TRANSPOSED for the WMMA B-side:
//   WnT[r][o][i]  = sum_b coeff[r,b]*basis[b][i][o]
//   WsSumT[o][i]  = sum_r W_self[r][i][o]
//   bias[o]       = sum_r b_rel[r][o] + h_bias[o]
// ---------------------------------------------------------------------------
__global__ void k_prep(const float* __restrict__ basis, const float* __restrict__ coeff,
                       const float* __restrict__ W_self, const float* __restrict__ b_rel,
                       const float* __restrict__ h_bias,
                       float* __restrict__ WnT, float* __restrict__ WsSumT,
                       float* __restrict__ biasS) {
    int idx = blockIdx.x * blockDim.x + threadIdx.x;   // over D*D, idx = i*D + o
    if (idx >= D * D) return;
    int i = idx >> 7;
    int o = idx & (D - 1);
    int tidx = o * D + i;                              // transposed slot
    float bv[B];
#pragma unroll
    for (int b = 0; b < B; ++b) bv[b] = basis[b * D * D + idx];
#pragma unroll
    for (int r = 0; r < R; ++r) {
        float acc = 0.f;
#pragma unroll
        for (int b = 0; b < B; ++b) acc += coeff[r * B + b] * bv[b];
        WnT[(size_t)r * D * D + tidx] = acc;
    }
    float s = 0.f;
#pragma unroll
    for (int r = 0; r < R; ++r) s += W_self[(size_t)r * D * D + idx];
    WsSumT[tidx] = s;
    if (idx < D) {
        float t = h_bias[idx];
#pragma unroll
        for (int r = 0; r < R; ++r) t += b_rel[r * D + idx];
        biasS[idx] = t;
    }
}

// ---------------------------------------------------------------------------
// x[n] = sum_f emb[f][codes[f][n]]   (float4 vectorized; emb is 2MB -> L2)
// ---------------------------------------------------------------------------
__global__ void k_gather(const int* __restrict__ codes, const float* __restrict__ emb,
                         float* __restrict__ x) {
    int t = blockIdx.x * blockDim.x + threadIdx.x;     // N * 32 threads
    if (t >= N * 32) return;
    int n = t >> 5;
    int d0 = (t & 31) * 4;
    float4 acc = make_float4(0.f, 0.f, 0.f, 0.f);
#pragma unroll
    for (int f = 0; f < NF; ++f) {
        int c = codes[f * N + n];
        const float4 e = *(const float4*)(emb + ((size_t)f * V + c) * D + d0);
        acc.x += e.x; acc.y += e.y; acc.z += e.z; acc.w += e.w;
    }
    *(float4*)(x + (size_t)n * D + d0) = acc;
}

// ---------------------------------------------------------------------------
// Zero a contiguous float region (msg followed by deg in ws)
// ---------------------------------------------------------------------------
__global__ void k_zero(float* __restrict__ p, int count) {
    int t = blockIdx.x * blockDim.x + threadIdx.x;
    if (t < count) p[t] = 0.f;
}

// ---------------------------------------------------------------------------
// Edge scatter: msg[dst] += x[src] ; deg[dst] += 1   (native f32 atomics)
// 32 consecutive lanes cover one 512B destination row -> coalesced atomics.
// ---------------------------------------------------------------------------
__global__ void k_scatter(const int* __restrict__ esrc, const int* __restrict__ edst,
                          const float* __restrict__ x, float* __restrict__ msg,
                          float* __restrict__ deg) {
    int t = blockIdx.x * blockDim.x + threadIdx.x;     // E * 32 threads
    if (t >= E * 32) return;
    int e = t >> 5;
    int c = t & 31;
    int d0 = c * 4;
    int s = esrc[e];
    int d = edst[e];
    const float4 v = *(const float4*)(x + (size_t)s * D + d0);
    float* mp = msg + (size_t)d * D + d0;
    __hip_atomic_fetch_add(mp + 0, v.x, __ATOMIC_RELAXED, __HIP_MEMORY_SCOPE_AGENT);
    __hip_atomic_fetch_add(mp + 1, v.y, __ATOMIC_RELAXED, __HIP_MEMORY_SCOPE_AGENT);
    __hip_atomic_fetch_add(mp + 2, v.z, __ATOMIC_RELAXED, __HIP_MEMORY_SCOPE_AGENT);
    __hip_atomic_fetch_add(mp + 3, v.w, __ATOMIC_RELAXED, __HIP_MEMORY_SCOPE_AGENT);
    if (c == 0)
        __hip_atomic_fetch_add(deg + d, 1.0f, __ATOMIC_RELAXED, __HIP_MEMORY_SCOPE_AGENT);
}

// ---------------------------------------------------------------------------
// Hacc (+)= diag(1/max(deg,1)) * (A @ W)  via V_WMMA_F32_16X16X4_F32.
//
// Block = 256 threads = 8 waves; block covers 16 rows x 128 cols of output;
// each wave owns one 16x16 tile, K=128 in steps of 4 (32 WMMAs).
//  - A tile rows 0-3 only (4x128 f32, 2KB) copied RAW into padded LDS with
//    GLOBAL_LOAD_ASYNC_TO_LDS_B128 (+ s_wait_asynccnt) when available.
//  - W is pre-transposed, so each lane's B-fragment is one b64 load and each
//    lane streams a contiguous 512B row of Wt over the K loop.
//  - degree normalization applied at the C store: diag(r)A@W = diag(r)(A@W).
//
// fp32 WMMA fragment layouts (ISA 7.12.2):
//   A 16x4 : lanes 0-15 -> M=lane, {v0,v1}={K0,K1}; lanes 16-31 -> {K2,K3}
//   B 4x16 : lanes 0-15 -> N=lane, {v0,v1}={K0,K1}; lanes 16-31 -> {K2,K3}
//   C 16x16: vgpr v, lanes 0-15 -> M=v, N=lane; lanes 16-31 -> M=v+8
// ---------------------------------------------------------------------------
__global__ __launch_bounds__(256)
void k_matmul(const float* __restrict__ A, const float* __restrict__ deg,
              const float* __restrict__ Wt, float* __restrict__ Hacc, int initC) {
    __shared__ float As[16][D + 4];    // +4 pad: conflict-free fragment reads
    __shared__ float rd_s[16];
    const int n0 = blockIdx.x * 16;
    const int t = threadIdx.x;

    if (t < 16) {
        float dg = deg ? deg[n0 + t] : 1.0f;
        rd_s[t] = 1.0f / fmaxf(dg, 1.0f);
    }

    // Stage A tile: 128 b128 chunks, one per thread of the first 4 waves
    if (t < 128) {
        int row = t >> 5;
        int col = (t & 31) * 4;
#if USE_ASYNC_LDS
        __builtin_amdgcn_global_load_async_to_lds_b128(
            (v4i*)(A + (size_t)(n0 + row) * D + col), (v4i*)&As[row][col], 0, 0);
#else
        *(float4*)&As[row][col] = *(const float4*)(A + (size_t)(n0 + row) * D + col);
#endif
    }
#if USE_ASYNC_LDS
    __builtin_amdgcn_s_wait_asynccnt(0);
#endif
    __syncthreads();

    const int wave = t >> 5;
    const int lane = t & 31;
    const int m  = lane & 15;    // row (A) / col (B,C) within tile
    const int kh = lane >> 4;    // K half selector
    const int col0 = wave * 16;

    const float* Wrow = Wt + (size_t)(col0 + m) * D;   // transposed weights

    v8f c = {};
#pragma unroll
    for (int k = 0; k < D; k += 4) {
        const int kk = k + 2 * kh;
        v2f a = *(const v2f*)&As[m][kk];       // ds_load_b64
        v2f b = *(const v2f*)(Wrow + kk);      // global_load_b64, streaming
        c = __builtin_amdgcn_wmma_f32_16x16x4_f32(false, a, false, b,
                                                  (short)0, c, false, false);
    }

    float* Hp = Hacc + (size_t)n0 * D + col0 + m;
#pragma unroll
    for (int v = 0; v < 8; ++v) {
        const int row = v + 8 * kh;
        float val = rd_s[row] * c[v];
        if (!initC) val += Hp[(size_t)row * D];
        Hp[(size_t)row * D] = val;
    }
}

// ---------------------------------------------------------------------------
// h = relu(Hacc + bias); logits = h @ W_cls + b_cls
// ---------------------------------------------------------------------------
__global__ void k_final(const float* __restrict__ Hacc, const float* __restrict__ biasS,
                        const float* __restrict__ W_cls, const float* __restrict__ b_cls,
                        float* __restrict__ out_logits, float* __restrict__ out_h) {
    __shared__ float hrow[D];
    const int n = blockIdx.x;
    const int d = threadIdx.x;   // 128
    float hv = Hacc[(size_t)n * D + d] + biasS[d];
    hv = fmaxf(hv, 0.0f);
    out_h[(size_t)n * D + d] = hv;
    hrow[d] = hv;
    __syncthreads();
    if (d < OUT) {
        float acc = b_cls[d];
#pragma unroll 8
        for (int i = 0; i < D; ++i) acc += hrow[i] * W_cls[i * OUT + d];
        out_logits[(size_t)n * OUT + d] = acc;
    }
}

// ---------------------------------------------------------------------------
extern "C" void kernel_launch(void* const* d_in, const int* in_sizes, int n_in,
                              void* d_out, int out_size, void* d_ws, size_t ws_size,
                              hipStream_t stream) {
    const int*   feat_codes = (const int*)d_in[0];
    // d_in[1] = nid = arange(N) -> codes = feat_codes, unused
    const int*   esrc   = (const int*)d_in[2];
    const int*   edst   = (const int*)d_in[3];
    const float* emb    = (const float*)d_in[4];
    const float* basis  = (const float*)d_in[5];
    const float* coeff  = (const float*)d_in[6];
    const float* W_self = (const float*)d_in[7];
    const float* b_rel  = (const float*)d_in[8];
    const float* h_bias = (const float*)d_in[9];
    const float* W_cls  = (const float*)d_in[10];
    const float* b_cls  = (const float*)d_in[11];
    float* out = (float*)d_out;

    // Workspace layout (floats): ~78 MB total
    float* ws    = (float*)d_ws;
    float* x     = ws;                       // N*D
    float* msg   = x + (size_t)N * D;        // N*D  (deg must follow msg!)
    float* deg   = msg + (size_t)N * D;      // N
    float* Hacc  = deg + N;                  // N*D
    float* WnT   = Hacc + (size_t)N * D;     // R*D*D (transposed)
    float* WsSumT= WnT + (size_t)R * D * D;  // D*D   (transposed)
    float* biasS = WsSumT + D * D;           // D

    // 1) fold weights (transposed for the WMMA B-side)
    k_prep<<<(D * D + 255) / 256, 256, 0, stream>>>(basis, coeff, W_self, b_rel,
                                                    h_bias, WnT, WsSumT, biasS);
    // 2) embedding gather-sum -> x
    k_gather<<<(N * 32 + 255) / 256, 256, 0, stream>>>(feat_codes, emb, x);
    // 3) self term initializes Hacc = x @ (sum_r W_self)
    k_matmul<<<N / 16, 256, 0, stream>>>(x, nullptr, WsSumT, Hacc, 1);
    // 4) per-relation: scatter-mean then Hacc += mean @ Wn[r]
    for (int r = 0; r < R; ++r) {
        k_zero<<<((N * D + N) + 255) / 256, 256, 0, stream>>>(msg, N * D + N);
        k_scatter<<<(E * 32 + 255) / 256, 256, 0, stream>>>(
            esrc + (size_t)r * E, edst + (size_t)r * E, x, msg, deg);
        k_matmul<<<N / 16, 256, 0, stream>>>(msg, deg, WnT + (size_t)r * D * D, Hacc, 0);
    }
    // 5) relu + bias + classifier -> d_out = [logits (N*OUT) | h (N*D)]
    k_final<<<N, D, 0, stream>>>(Hacc, biasS, W_cls, b_cls, out, out + (size_t)N * OUT);
}